// QwenDecoder_20658792694698
// MI455X (gfx1250) — compile-verified
//
#include <hip/hip_runtime.h>
#include <math.h>

#define LAYERS 28
#define NHQ 16
#define NKVH 8
#define HD 128
#define HDIM 1024
#define FFDIM 3072
#define VOCAB 32768
#define SLEN 1024
#define EPSF 1e-6f

typedef __attribute__((ext_vector_type(2))) float v2f;
typedef __attribute__((ext_vector_type(8))) float v8f;

__device__ __forceinline__ float wave_sum(float v) {
  for (int off = 16; off; off >>= 1) v += __shfl_xor(v, off, 32);
  return v;
}
__device__ __forceinline__ float wave_max(float v) {
  for (int off = 16; off; off >>= 1) v = fmaxf(v, __shfl_xor(v, off, 32));
  return v;
}

// Block-wide 1/rms over x[0..K). redbuf must have blockDim.x/32 slots.
__device__ __forceinline__ float block_rstd(const float* __restrict__ x, int K,
                                            float* redbuf) {
  int tid = threadIdx.x, lane = tid & 31, wave = tid >> 5;
  float ss = 0.f;
  for (int i = tid; i < K; i += blockDim.x) { float v = x[i]; ss += v * v; }
  ss = wave_sum(ss);
  if (!lane) redbuf[wave] = ss;
  __syncthreads();
  int nwv = blockDim.x >> 5;
  float tot = 0.f;
  for (int w = 0; w < nwv; ++w) tot += redbuf[w];
  return rsqrtf(tot / (float)K + EPSF);
}

// ---------------- embedding gather ----------------
__global__ void gather_kernel(const int* __restrict__ ids,
                              const float* __restrict__ embed,
                              float* __restrict__ h) {
  int id = ids[0];
  for (int i = threadIdx.x; i < HDIM; i += blockDim.x)
    h[i] = embed[(size_t)id * HDIM + i];
}

// ---------------- fused rmsnorm + q/k/v projections ----------------
__global__ __launch_bounds__(256) void qkv_kernel(
    const float* __restrict__ x, const float* __restrict__ nw,
    const float* __restrict__ qw, const float* __restrict__ qb,
    const float* __restrict__ kw, const float* __restrict__ kb,
    const float* __restrict__ vw, const float* __restrict__ vb,
    float* __restrict__ out) {
  __shared__ __align__(16) float xs[HDIM];
  __shared__ float red[8];
  int tid = threadIdx.x, lane = tid & 31, wave = tid >> 5;
  float rstd = block_rstd(x, HDIM, red);
  for (int i = tid; i < HDIM; i += 256) xs[i] = x[i] * rstd * nw[i];
  __syncthreads();
  int r = blockIdx.x * 8 + wave;
  const float* W; const float* B; int o;
  if (r < 2048)      { W = qw; B = qb; o = r; }
  else if (r < 3072) { W = kw; B = kb; o = r - 2048; }
  else               { W = vw; B = vb; o = r - 3072; }
  const float* Wr = W + (size_t)o * HDIM;
  float acc = 0.f;
  for (int i0 = lane * 4; i0 < HDIM; i0 += 128) {
    float4 w4 = *(const float4*)(Wr + i0);
    float4 x4 = *(const float4*)(xs + i0);
    acc = fmaf(w4.x, x4.x, acc); acc = fmaf(w4.y, x4.y, acc);
    acc = fmaf(w4.z, x4.z, acc); acc = fmaf(w4.w, x4.w, acc);
  }
  acc = wave_sum(acc);
  if (!lane) out[r] = acc + B[o];
}

// ---------------- generic gemv: y[o] = (res?) + dot(norm(x), W[o,:]) --------
__global__ __launch_bounds__(256) void gemv_kernel(
    const float* __restrict__ x, int K,
    const float* __restrict__ nw,     // nullable: rmsnorm weight
    const float* __restrict__ W,
    const float* __restrict__ res,    // nullable: residual add
    float* __restrict__ y) {
  __shared__ __align__(16) float xs[FFDIM];
  __shared__ float red[8];
  int tid = threadIdx.x, lane = tid & 31, wave = tid >> 5;
  float rstd = 1.f;
  if (nw) rstd = block_rstd(x, K, red);
  for (int i = tid; i < K; i += 256) xs[i] = nw ? x[i] * rstd * nw[i] : x[i];
  __syncthreads();
  int o = blockIdx.x * 8 + wave;
  const float* Wr = W + (size_t)o * K;
  float acc = 0.f;
  for (int i0 = lane * 4; i0 < K; i0 += 128) {
    float4 w4 = *(const float4*)(Wr + i0);
    float4 x4 = *(const float4*)(xs + i0);
    acc = fmaf(w4.x, x4.x, acc); acc = fmaf(w4.y, x4.y, acc);
    acc = fmaf(w4.z, x4.z, acc); acc = fmaf(w4.w, x4.w, acc);
  }
  acc = wave_sum(acc);
  if (!lane) y[o] = res ? res[o] + acc : acc;
}

// ---------------- fused rmsnorm + gate/up + SiLU ----------------
__global__ __launch_bounds__(256) void gateup_kernel(
    const float* __restrict__ x, const float* __restrict__ nw,
    const float* __restrict__ gw, const float* __restrict__ uw,
    float* __restrict__ t) {
  __shared__ __align__(16) float xs[HDIM];
  __shared__ float red[8];
  int tid = threadIdx.x, lane = tid & 31, wave = tid >> 5;
  float rstd = block_rstd(x, HDIM, red);
  for (int i = tid; i < HDIM; i += 256) xs[i] = x[i] * rstd * nw[i];
  __syncthreads();
  int o = blockIdx.x * 8 + wave;
  const float* Gr = gw + (size_t)o * HDIM;
  const float* Ur = uw + (size_t)o * HDIM;
  float ag = 0.f, au = 0.f;
  for (int i0 = lane * 4; i0 < HDIM; i0 += 128) {
    float4 x4 = *(const float4*)(xs + i0);
    float4 g4 = *(const float4*)(Gr + i0);
    float4 u4 = *(const float4*)(Ur + i0);
    ag = fmaf(g4.x, x4.x, ag); ag = fmaf(g4.y, x4.y, ag);
    ag = fmaf(g4.z, x4.z, ag); ag = fmaf(g4.w, x4.w, ag);
    au = fmaf(u4.x, x4.x, au); au = fmaf(u4.y, x4.y, au);
    au = fmaf(u4.z, x4.z, au); au = fmaf(u4.w, x4.w, au);
  }
  ag = wave_sum(ag); au = wave_sum(au);
  if (!lane) t[o] = (ag / (1.f + expf(-ag))) * au;
}

// ------- per-head q/k rmsnorm + rope; scatter k,v rows into output cache ----
__global__ __launch_bounds__(128) void rope_cache_kernel(
    const float* __restrict__ qkv, const float* __restrict__ qnw,
    const float* __restrict__ knw, const int* __restrict__ posp,
    float* __restrict__ qout, float* __restrict__ kvout, int layer) {
  __shared__ float xn[HD];
  __shared__ float red4[4];
  int b = blockIdx.x;          // 0-15 q heads, 16-23 k heads, 24-31 v heads
  int d = threadIdx.x;         // 0..127
  int lane = d & 31, wave = d >> 5;
  int pos = posp[0];
  if (b >= 24) {               // v heads: plain copy into cache
    int j = b - 24;
    float v = qkv[3072 + j * HD + d];
    kvout[((size_t)(2 * layer + 1) * NKVH + j) * SLEN * HD + (size_t)pos * HD + d] = v;
    return;
  }
  bool isq = (b < 16);
  const float* src = isq ? (qkv + b * HD) : (qkv + 2048 + (b - 16) * HD);
  const float* w   = isq ? qnw : knw;
  float v = src[d];
  float ss = wave_sum(v * v);
  if (!lane) red4[wave] = ss;
  __syncthreads();
  float rstd = rsqrtf((red4[0] + red4[1] + red4[2] + red4[3]) / (float)HD + EPSF);
  float nrm = v * rstd * w[d];
  xn[d] = nrm;
  __syncthreads();
  float partner = (d < 64) ? -xn[d + 64] : xn[d - 64];
  int i = d & 63;
  // inv_freq = theta^(-i/64), theta = 1e6
  float inv = expf(-(float)i * (13.815510557964274f / 64.f));
  float ang = (float)pos * inv;
  float out = nrm * cosf(ang) + partner * sinf(ang);
  if (isq) qout[b * HD + d] = out;
  else {
    int j = b - 16;
    kvout[((size_t)(2 * layer) * NKVH + j) * SLEN * HD + (size_t)pos * HD + d] = out;
  }
}

// ------- attention: WMMA f32 16x16x4 scores + softmax + P.V ----------------
__global__ __launch_bounds__(1024) void attn_kernel(
    const float* __restrict__ q,      // 16 x 128 post-rope
    const float* __restrict__ kvout,  // output kv cache base
    const int* __restrict__ posp,
    float* __restrict__ ao,           // 16 x 128
    int layer) {
  const int j = blockIdx.x;           // kv head
  const int tid = threadIdx.x;        // 0..1023
  const int lane = tid & 31;
  const int wave = tid >> 5;          // 0..31
  const int pos = posp[0];
  const float* Kc = kvout + ((size_t)(2 * layer) * NKVH + j) * SLEN * HD;
  const float* Vc = kvout + ((size_t)(2 * layer + 1) * NKVH + j) * SLEN * HD;

  __shared__ float qs[2][HD];
  __shared__ float sc[2][SLEN];
  __shared__ float red[32];
  __shared__ float part[8][HD];

  for (int i = tid; i < 2 * HD; i += 1024) qs[i >> 7][i & 127] = q[2 * j * HD + i];
  __syncthreads();

  const float scale = 0.08838834764831845f;  // 1/sqrt(128)

  // Scores: D = A(16 cache rows x 4) * B(4 x 16; cols 0/1 = the two GQA q heads)
  // chained over K=128 via v_wmma_f32_16x16x4_f32 (full f32 precision).
  const int m = lane & 15;
  const int khalf = (lane >> 4) << 1;            // 0 or 2 (A/B vgpr layout)
  const float* qcol = ((lane & 15) == 1) ? qs[1] : qs[0];
  for (int c = wave; c < 64; c += 32) {
    int s0 = c * 16;
    const float* krow = Kc + (size_t)(s0 + m) * HD + khalf;
    v8f acc = {};
    for (int kb = 0; kb < HD; kb += 4) {
      float2 av = *(const float2*)(krow + kb);
      float2 bv = *(const float2*)(qcol + kb + khalf);
      v2f a, b;
      a.x = av.x; a.y = av.y;
      b.x = bv.x; b.y = bv.y;
      acc = __builtin_amdgcn_wmma_f32_16x16x4_f32(false, a, false, b,
                                                  (short)0, acc, false, false);
    }
    int n = lane & 15;
    if (n < 2) {
      int mbase = (lane >> 4) << 3;
      for (int r = 0; r < 8; ++r) {
        int s = s0 + mbase + r;
        float val = acc[r] * scale;
        sc[n][s] = (s <= pos) ? val : -INFINITY;
      }
    }
  }
  __syncthreads();

  // softmax per head (block-wide over SLEN)
  for (int hh = 0; hh < 2; ++hh) {
    float v = sc[hh][tid];
    float mx = wave_max(v);
    if (!lane) red[wave] = mx;
    __syncthreads();
    float gm = wave_max(red[lane]);
    float e = expf(v - gm);
    float s = wave_sum(e);
    __syncthreads();
    if (!lane) red[wave] = s;
    __syncthreads();
    float gs = wave_sum(red[lane]);
    sc[hh][tid] = e / gs;
    __syncthreads();
  }

  // output: ao[head][d] = sum_s p[s] * V[s][d]  (coalesced V rows)
  int d = tid & 127;
  int g = tid >> 7;       // 0..7
  int hh = g & 1;
  int sr = g >> 1;        // 0..3
  float accv = 0.f;
  const float* Vp = Vc + d;
  for (int s = sr * 256; s < sr * 256 + 256; ++s)
    accv = fmaf(sc[hh][s], Vp[(size_t)s * HD], accv);
  part[g][d] = accv;
  __syncthreads();
  if (tid < 256) {
    int h2 = tid >> 7, d2 = tid & 127;
    float o = part[h2][d2] + part[h2 + 2][d2] + part[h2 + 4][d2] + part[h2 + 6][d2];
    ao[(2 * j + h2) * HD + d2] = o;
  }
}

extern "C" void kernel_launch(void* const* d_in, const int* in_sizes, int n_in,
                              void* d_out, int out_size, void* d_ws, size_t ws_size,
                              hipStream_t stream) {
  (void)in_sizes; (void)n_in; (void)out_size; (void)ws_size;
  const int*   ids   = (const int*)d_in[0];
  const int*   posp  = (const int*)d_in[1];
  const float* kvin  = (const float*)d_in[2];
  const float* embed = (const float*)d_in[3];
  const float* fnw   = (const float*)d_in[4];
  const float* ln1   = (const float*)d_in[5];
  const float* ln2   = (const float*)d_in[6];
  const float* qw    = (const float*)d_in[7];
  const float* qb    = (const float*)d_in[8];
  const float* kw    = (const float*)d_in[9];
  const float* kb    = (const float*)d_in[10];
  const float* vw    = (const float*)d_in[11];
  const float* vb    = (const float*)d_in[12];
  const float* ow    = (const float*)d_in[13];
  const float* qnw   = (const float*)d_in[14];
  const float* knw   = (const float*)d_in[15];
  const float* gw    = (const float*)d_in[16];
  const float* uw    = (const float*)d_in[17];
  const float* dw    = (const float*)d_in[18];

  float* out    = (float*)d_out;
  float* logits = out;
  float* kvout  = out + VOCAB;

  float* ws  = (float*)d_ws;
  float* h   = ws;            // 1024
  float* qkv = ws + 1024;     // 4096 (q | k | v)
  float* qpr = ws + 5120;     // 2048 post-rope q
  float* ao  = ws + 7168;     // 2048 attention out
  float* t   = ws + 9216;     // 3072 mlp activation

  // KV cache passthrough into the output buffer (updated rows written below).
  hipMemcpyAsync(kvout, kvin,
                 (size_t)2 * LAYERS * NKVH * SLEN * HD * sizeof(float),
                 hipMemcpyDeviceToDevice, stream);

  gather_kernel<<<1, 256, 0, stream>>>(ids, embed, h);

  for (int l = 0; l < LAYERS; ++l) {
    qkv_kernel<<<512, 256, 0, stream>>>(
        h, ln1 + (size_t)l * HDIM,
        qw + (size_t)l * 2048 * HDIM, qb + (size_t)l * 2048,
        kw + (size_t)l * 1024 * HDIM, kb + (size_t)l * 1024,
        vw + (size_t)l * 1024 * HDIM, vb + (size_t)l * 1024, qkv);
    rope_cache_kernel<<<32, 128, 0, stream>>>(
        qkv, qnw + (size_t)l * HD, knw + (size_t)l * HD, posp, qpr, kvout, l);
    attn_kernel<<<NKVH, 1024, 0, stream>>>(qpr, kvout, posp, ao, l);
    gemv_kernel<<<128, 256, 0, stream>>>(
        ao, 2048, nullptr, ow + (size_t)l * HDIM * 2048, h, h);
    gateup_kernel<<<384, 256, 0, stream>>>(
        h, ln2 + (size_t)l * HDIM,
        gw + (size_t)l * FFDIM * HDIM, uw + (size_t)l * FFDIM * HDIM, t);
    gemv_kernel<<<128, 256, 0, stream>>>(
        t, FFDIM, nullptr, dw + (size_t)l * HDIM * FFDIM, h, h);
  }

  // final rmsnorm + logits over the vocabulary
  gemv_kernel<<<VOCAB / 8, 256, 0, stream>>>(
      h, HDIM, fnw, embed, nullptr, logits);
}